// Sum_47562467836378
// MI455X (gfx1250) — compile-verified
//
#include <hip/hip_runtime.h>
#include <math.h>

// Problem constants (from reference): N=256, W=1, D=64, IC=32, OC=32, R=8
// x   [n, 0, d, ic, r] : idx = n*16384 + d*256 + ic*8 + r
// w   [d, ic, oc, r]   : idx = d*8192  + ic*256 + oc*8 + r
// out [n, 0, d, oc, r] : idx = n*16384 + d*256 + oc*8 + r

typedef float v2f __attribute__((ext_vector_type(2)));
typedef float v8f __attribute__((ext_vector_type(8)));

#define BROW 40  // padded LDS row stride (dwords): rows K and K+2 differ by 80 -> +16 banks mod 64,
                 // so lanes 0-15 (row K) and 16-31 (row K+2) hit disjoint 16-bank groups.

// Fast transcendentals: exactly one v_exp_f32 / v_log_f32 + one v_mul each.
// exp(x) = exp2(x * log2(e)); log(x) = log2(x) * ln(2).
__device__ __forceinline__ float fast_exp(float x) {
    return __builtin_amdgcn_exp2f(x * 1.4426950408889634f);
}
__device__ __forceinline__ float fast_log(float x) {
    return __builtin_amdgcn_logf(x) * 0.6931471805599453f;
}

__global__ __launch_bounds__(256) void lse_wmma_kernel(const float* __restrict__ x,
                                                       const float* __restrict__ w,
                                                       float* __restrict__ out) {
    const int d = blockIdx.x >> 3;   // 0..63
    const int r = blockIdx.x & 7;    // 0..7

    __shared__ float sB[32 * BROW];  // exp(w - mw), [ic][oc] padded
    __shared__ float sMw[32];        // per-oc max over ic
    __shared__ float sPmax[8 * 32];  // partial maxes

    const int tid   = threadIdx.x;
    const int oc    = tid & 31;
    const int chunk = tid >> 5;      // 0..7 -> ic group of 4

    // ---- Phase 1: B = exp(w - colmax(w)) into LDS ----
    const float* wp = w + (size_t)d * 8192 + (size_t)oc * 8 + r;
    float wv[4];
    float pm = -__builtin_inff();
#pragma unroll
    for (int j = 0; j < 4; ++j) {
        const int ic = chunk * 4 + j;
        wv[j] = wp[(size_t)ic * 256];
        pm = fmaxf(pm, wv[j]);
    }
    sPmax[chunk * 32 + oc] = pm;
    __syncthreads();
    if (tid < 32) {
        float m = sPmax[tid];
#pragma unroll
        for (int c = 1; c < 8; ++c) m = fmaxf(m, sPmax[c * 32 + tid]);
        sMw[tid] = m;
    }
    __syncthreads();
    {
        const float m = sMw[oc];
#pragma unroll
        for (int j = 0; j < 4; ++j) {
            const int ic = chunk * 4 + j;
            sB[ic * BROW + oc] = fast_exp(wv[j] - m);
        }
    }
    __syncthreads();

    // ---- Phase 2: per-wave 16x16 output tiles via V_WMMA_F32_16X16X4_F32 ----
    const int lane = tid & 31;
    const int wave = tid >> 5;       // 0..7
    const int half = lane >> 4;      // 0 | 1
    const int Mr   = lane & 15;      // row-in-tile (A) / col-in-tile (B,C)

    for (int mt = wave; mt < 16; mt += 8) {           // 2 M-tiles per wave, uniform
        const int n0 = mt * 16;
        // A layout (ISA 16x4 f32): lane<16 holds K=4k+{0,1} of row Mr; lane>=16 holds K=4k+{2,3}
        const float* px = x + (size_t)(n0 + Mr) * 16384 + (size_t)d * 256 + r + half * 16;

        float av[16];
        float mymax = -__builtin_inff();
#pragma unroll
        for (int k = 0; k < 8; ++k) {
            const float v0 = px[k * 32];
            const float v1 = px[k * 32 + 8];
            av[2 * k]     = v0;
            av[2 * k + 1] = v1;
            mymax = fmaxf(mymax, fmaxf(v0, v1));
        }
        // exact row max: combine the two lane-halves (together they cover all 32 ic)
        const float full = fmaxf(mymax, __shfl_xor(mymax, 16, 32));

        v2f afrag[8];
#pragma unroll
        for (int k = 0; k < 8; ++k) {
            afrag[k].x = fast_exp(av[2 * k]     - full);
            afrag[k].y = fast_exp(av[2 * k + 1] - full);
        }

        // redistribute row maxes into C/D layout: VGPR v holds row M = v + 8*half
        float mrow[8];
#pragma unroll
        for (int v = 0; v < 8; ++v)
            mrow[v] = __shfl(full, v + 8 * half, 32);

#pragma unroll
        for (int nt = 0; nt < 2; ++nt) {
            v8f acc = {};
#pragma unroll
            for (int k = 0; k < 8; ++k) {
                // B layout (4x16 f32, mirrors C/D striping): VGPR0 = rows {4k, 4k+2}, VGPR1 = rows {4k+1, 4k+3}
                const int row = 4 * k + 2 * half;
                v2f b;
                b.x = sB[row * BROW       + nt * 16 + Mr];
                b.y = sB[(row + 1) * BROW + nt * 16 + Mr];
                acc = __builtin_amdgcn_wmma_f32_16x16x4_f32(
                    /*neg_a=*/false, afrag[k], /*neg_b=*/false, b,
                    /*c_mod=*/(short)0, acc, /*reuse_a=*/false, /*reuse_b=*/false);
            }
            const float mwv  = sMw[nt * 16 + Mr];
            float* pout = out + (size_t)(n0 + 8 * half) * 16384 + (size_t)d * 256
                              + (size_t)(nt * 16 + Mr) * 8 + r;
#pragma unroll
            for (int v = 0; v < 8; ++v) {
                const float s = acc[v];
                pout[(size_t)v * 16384] = fast_log(fmaxf(s, 1e-38f)) + mrow[v] + mwv;
            }
        }
    }
}

extern "C" void kernel_launch(void* const* d_in, const int* in_sizes, int n_in,
                              void* d_out, int out_size, void* d_ws, size_t ws_size,
                              hipStream_t stream) {
    const float* x = (const float*)d_in[0];   // [256,1,64,32,8] f32
    const float* w = (const float*)d_in[1];   // [64,32,32,8]    f32
    float* out = (float*)d_out;               // [256,1,64,32,8] f32

    dim3 grid(64 * 8);    // one block per (d, r)
    dim3 block(256);      // 8 wave32 waves
    hipLaunchKernelGGL(lse_wmma_kernel, grid, block, 0, stream, x, w, out);
}